// MAD_GCN_23081154248747
// MI455X (gfx1250) — compile-verified
//
#include <hip/hip_runtime.h>
#include <hip/hip_bf16.h>

typedef __attribute__((ext_vector_type(2))) float v2f;
typedef __attribute__((ext_vector_type(4))) float v4f;
typedef __attribute__((ext_vector_type(8))) float v8f;

// Problem constants (from the reference)
constexpr int N_ = 16384;
constexpr int H_ = 4;
constexpr int D_ = 64;
constexpr int B_ = 512;
constexpr int K_ = 8;
constexpr int NQ = 2 * H_ * B_;                 // 4096 queries: q = side*H*B + h*B + b

constexpr int TQ  = 16;                         // queries per workgroup (one WMMA tile M)
constexpr int TQP = 17;                         // padded LDS row to dodge bank conflicts
constexpr int CH  = 128;                        // nodes per chunk (8 waves x 16)
constexpr int NCH = N_ / CH;                    // 128 chunks
constexpr int SPQ = 16;                         // selection threads per query

// ---------------------------------------------------------------------------
// sum of squares over contiguous 64-float rows (used for ||pos||^2 and ||q||^2)
// ---------------------------------------------------------------------------
__global__ void rowsumsq_k(const float* __restrict__ src, float* __restrict__ dst, int rows) {
  int r = blockIdx.x * blockDim.x + threadIdx.x;
  if (r >= rows) return;
  const v4f* p = (const v4f*)(src + (size_t)r * D_);
  float s = 0.f;
#pragma unroll
  for (int i = 0; i < D_ / 4; ++i) {
    v4f v = p[i];
    s += v.x * v.x + v.y * v.y + v.z * v.z + v.w * v.w;
  }
  dst[r] = s;
}

// ---------------------------------------------------------------------------
// gather query positions and (swapped) query gradients into dense tables
// q = side*H*B + h*B + b ; side 0 queries pos[src], uses grads[dst]; side 1 vice versa
// ---------------------------------------------------------------------------
__global__ void gather_k(const float* __restrict__ pos, const float* __restrict__ grads,
                         const int* __restrict__ edges,
                         float* __restrict__ Qpos, float* __restrict__ Qgrad) {
  int t = blockIdx.x * blockDim.x + threadIdx.x;
  if (t >= NQ * D_) return;
  int q = t >> 6, d = t & 63;
  int side = q / (H_ * B_);
  int rem  = q - side * (H_ * B_);
  int h = rem / B_, b = rem - h * B_;
  int node  = edges[side * B_ + b];
  int other = edges[(1 - side) * B_ + b];
  Qpos[t]  = pos[((size_t)node * H_ + h) * D_ + d];
  Qgrad[t] = grads[((size_t)other * H_ + h) * D_ + d];
}

// ---------------------------------------------------------------------------
// KNN: per workgroup, 16 queries vs all N nodes.
//  - 8 waves each compute a 16x16 f32 WMMA tile (K-loop over D=64 in steps of 4)
//  - B fragments double-buffered: chunk c+1 loads fly during chunk c's WMMA
//    chain and the selection scan, so WMMAs never stall on a VMEM round-trip
//  - dist^2 = pn2 + q2 - 2*cross written to LDS
//  - 256 threads (16 per query) stream a lexicographic top-9, then merge
// ---------------------------------------------------------------------------
__launch_bounds__(256)
__global__ void knn_k(const float* __restrict__ pos, const float* __restrict__ pn2,
                      const float* __restrict__ Qpos, const float* __restrict__ q2,
                      int* __restrict__ knnIdx) {
  __shared__ float buf[CH * TQP];          // dist^2 chunk, padded rows
  __shared__ float mv[TQ * SPQ * 9];       // partial top-9 values
  __shared__ int   mi[TQ * SPQ * 9];       // partial top-9 indices

  const int tid  = threadIdx.x;
  const int lane = tid & 31;
  const int wave = tid >> 5;
  const int hi   = lane >> 4;              // lane-half (ISA 16x4 layout)
  const int lo   = lane & 15;
  const int qbase = blockIdx.x * TQ;
  const int h = (qbase / B_) % H_;         // all 16 queries share (side, head)

  // Resident A fragments: A[m][k] with m=lo, a.x=K(k0+2*hi), a.y=K(k0+2*hi+1)
  v2f aF[16];
#pragma unroll
  for (int ks = 0; ks < 16; ++ks)
    aF[ks] = *(const v2f*)(Qpos + (size_t)(qbase + lo) * D_ + ks * 4 + 2 * hi);

  // q^2 for the 8 C-rows this lane owns (rows m = 8*hi + r)
  float q2v[8];
#pragma unroll
  for (int r = 0; r < 8; ++r) q2v[r] = q2[qbase + 8 * hi + r];

  // Streaming top-9 state (ascending, lexicographic (val, idx))
  const int m_sel = tid >> 4;
  const int s_sel = tid & 15;
  float vals[9];
  int   idxs[9];
#pragma unroll
  for (int e = 0; e < 9; ++e) { vals[e] = __builtin_inff(); idxs[e] = 0x7FFFFFFF; }

  // --- double-buffered B fragments + pn2 ---
  v2f bcur[16], bnext[16];
  float pncur, pnnext;
  {
    const int nodeLane = wave * TQ + lo;   // chunk 0
    const float* Pb = pos + ((size_t)nodeLane * H_ + h) * D_ + 2 * hi;
#pragma unroll
    for (int ks = 0; ks < 16; ++ks) bcur[ks] = *(const v2f*)(Pb + ks * 4);
    pncur = pn2[(size_t)nodeLane * H_ + h];
  }

  for (int c = 0; c < NCH; ++c) {
    // prefetch chunk c+1 (independent of the WMMA chain below)
    if (c + 1 < NCH) {
      const int nodeLane = (c + 1) * CH + wave * TQ + lo;
      const float* Pb = pos + ((size_t)nodeLane * H_ + h) * D_ + 2 * hi;
#pragma unroll
      for (int ks = 0; ks < 16; ++ks) bnext[ks] = *(const v2f*)(Pb + ks * 4);
      pnnext = pn2[(size_t)nodeLane * H_ + h];
    }

    // --- 16q x 16n cross-product tile via f32 WMMA (operands resident) ---
    v8f acc = {};
#pragma unroll
    for (int ks = 0; ks < 16; ++ks) {
      acc = __builtin_amdgcn_wmma_f32_16x16x4_f32(
          /*neg_a=*/false, aF[ks], /*neg_b=*/false, bcur[ks],
          /*c_mod=*/(short)0, acc, /*reuse_a=*/false, /*reuse_b=*/false);
    }

#pragma unroll
    for (int r = 0; r < 8; ++r) {
      const int m = r + 8 * hi;                      // C row (query)
      buf[(wave * TQ + lo) * TQP + m] = pncur + q2v[r] - 2.0f * acc[r];
    }
    __syncthreads();

    // --- parallel streaming selection: thread (m_sel, s_sel) scans 8 nodes ---
#pragma unroll
    for (int j = 0; j < CH / SPQ; ++j) {
      const int nn = s_sel * (CH / SPQ) + j;
      const float d = buf[nn * TQP + m_sel];
      const int   gi = c * CH + nn;
      if (d < vals[8] || (d == vals[8] && gi < idxs[8])) {
        vals[8] = d; idxs[8] = gi;
#pragma unroll
        for (int e = 8; e >= 1; --e) {
          float ve = vals[e], vp = vals[e - 1];
          int   ie = idxs[e], ip = idxs[e - 1];
          bool sw = (ve < vp) || (ve == vp && ie < ip);
          vals[e]     = sw ? vp : ve;
          vals[e - 1] = sw ? ve : vp;
          idxs[e]     = sw ? ip : ie;
          idxs[e - 1] = sw ? ie : ip;
        }
      }
    }
    __syncthreads();

    // rotate buffers (renamed away by the compiler after unrolling)
#pragma unroll
    for (int ks = 0; ks < 16; ++ks) bcur[ks] = bnext[ks];
    pncur = pnnext;
  }

  // --- dump partial lists and merge (one thread per query) ---
  {
    const int base = (m_sel * SPQ + s_sel) * 9;
#pragma unroll
    for (int e = 0; e < 9; ++e) { mv[base + e] = vals[e]; mi[base + e] = idxs[e]; }
  }
  __syncthreads();

  if (tid < TQ) {
    const int m = tid;
    const int base = m * SPQ * 9;
    for (int out = 0; out < 9; ++out) {
      float bv = __builtin_inff(); int bi = 0x7FFFFFFF; int bp = 0;
      for (int e2 = 0; e2 < SPQ * 9; ++e2) {
        float v = mv[base + e2]; int ii = mi[base + e2];
        if (v < bv || (v == bv && ii < bi)) { bv = v; bi = ii; bp = e2; }
      }
      mv[base + bp] = __builtin_inff(); mi[base + bp] = 0x7FFFFFFF;
      if (out >= 1)                                   // drop rank-0 (self)
        knnIdx[(size_t)(qbase + m) * K_ + (out - 1)] = bi;
    }
  }
}

// ---------------------------------------------------------------------------
// per (b,h): neighbor diffs, adj gathers, softmax-with-sentinels, partial sum
// ---------------------------------------------------------------------------
__global__ void finalize_k(const float* __restrict__ pos, const float* __restrict__ Qpos,
                           const float* __restrict__ Qgrad, const float* __restrict__ adj,
                           const float* __restrict__ label_w, const int* __restrict__ edges,
                           const int* __restrict__ knnIdx, float* __restrict__ hsum) {
  int t = blockIdx.x * blockDim.x + threadIdx.x;
  if (t >= B_ * H_) return;
  int b = t / H_, h = t - b * H_;
  int src_b = edges[b], dst_b = edges[B_ + b];
  float lw = label_w[0];

  float dists[16], logits[16];
#pragma unroll
  for (int side = 0; side < 2; ++side) {
    const int q = side * (H_ * B_) + h * B_ + b;
    const float* pq = Qpos  + (size_t)q * D_;
    const float* gq = Qgrad + (size_t)q * D_;
#pragma unroll
    for (int k = 0; k < K_; ++k) {
      int nbr = knnIdx[(size_t)q * K_ + k];
      const float* pn = pos + ((size_t)nbr * H_ + h) * D_;
      float dot = 0.f, nr2 = 0.f;
#pragma unroll 4
      for (int d = 0; d < D_; ++d) {
        float df = pq[d] - pn[d];
        dot = fmaf(df, gq[d], dot);
        nr2 = fmaf(df, df, nr2);
      }
      float adjv = (side == 0) ? adj[(size_t)nbr * N_ + dst_b]
                               : adj[(size_t)src_b * N_ + nbr];
      dists[side * K_ + k]  = sqrtf(nr2);
      logits[side * K_ + k] = lw * adjv + dot;
    }
  }

  // softmax(-dist) over 16 real + 8 sentinel (dist=1, logit=0) slots
  float mx = -1.0f;
#pragma unroll
  for (int i = 0; i < 16; ++i) mx = fmaxf(mx, -dists[i]);
  float Z = 8.0f * expf(-1.0f - mx);
  float sa = 0.f;
#pragma unroll
  for (int i = 0; i < 16; ++i) {
    float e = expf(-dists[i] - mx);
    Z += e;
    sa += e * logits[i];
  }
  hsum[t] = sa / Z;
}

__global__ void pred_k(const float* __restrict__ hsum, float* __restrict__ out) {
  int b = blockIdx.x * blockDim.x + threadIdx.x;
  if (b >= B_) return;
  float s = 0.f;
#pragma unroll
  for (int h = 0; h < H_; ++h) s += hsum[b * H_ + h];
  s *= (1.0f / H_);
  out[b] = 1.0f / (1.0f + expf(-s));
}

// ---------------------------------------------------------------------------
extern "C" void kernel_launch(void* const* d_in, const int* in_sizes, int n_in,
                              void* d_out, int out_size, void* d_ws, size_t ws_size,
                              hipStream_t stream) {
  const float* pos     = (const float*)d_in[0];
  const float* grads   = (const float*)d_in[1];
  const float* adj     = (const float*)d_in[2];
  const float* label_w = (const float*)d_in[3];
  const int*   edges   = (const int*)d_in[4];
  float* out = (float*)d_out;

  char* ws = (char*)d_ws;
  float* pn2    = (float*)(ws);                                       // N*H      = 256 KB
  float* Qpos   = (float*)(ws + 262144);                              // NQ*D     = 1 MB
  float* Qgrad  = (float*)(ws + 262144 + 1048576);                    // NQ*D     = 1 MB
  float* q2     = (float*)(ws + 262144 + 2097152);                    // NQ       = 16 KB
  int*   knnIdx = (int*)  (ws + 262144 + 2097152 + 16384);            // NQ*K     = 128 KB
  float* hsum   = (float*)(ws + 262144 + 2097152 + 16384 + 131072);   // B*H      = 8 KB

  rowsumsq_k<<<(N_ * H_ + 255) / 256, 256, 0, stream>>>(pos, pn2, N_ * H_);
  gather_k  <<<(NQ * D_ + 255) / 256, 256, 0, stream>>>(pos, grads, edges, Qpos, Qgrad);
  rowsumsq_k<<<(NQ + 255) / 256, 256, 0, stream>>>(Qpos, q2, NQ);
  knn_k     <<<NQ / TQ, 256, 0, stream>>>(pos, pn2, Qpos, q2, knnIdx);
  finalize_k<<<(B_ * H_ + 255) / 256, 256, 0, stream>>>(pos, Qpos, Qgrad, adj, label_w,
                                                        edges, knnIdx, hsum);
  pred_k    <<<(B_ + 255) / 256, 256, 0, stream>>>(hsum, out);
}